// gMPNN_46042049413371
// MI455X (gfx1250) — compile-verified
//
#include <hip/hip_runtime.h>

#define NN 1024   // nodes
#define HH 64     // hidden
#define EE 32768  // edges
#define NG 32     // graphs
#define RR 64     // rows of hid we actually need (src ∪ dst)

typedef __attribute__((ext_vector_type(2))) float v2f;
typedef __attribute__((ext_vector_type(8))) float v8f;

__global__ void k_zero(float* __restrict__ p, int n) {
    int i = blockIdx.x * blockDim.x + threadIdx.x;
    if (i < n) p[i] = 0.0f;
}

// A[e0, e1] += 1  (duplicate edges accumulate, matching coalesce semantics)
__global__ void k_build_A(const int* __restrict__ ei, float* __restrict__ A) {
    int e = blockIdx.x * blockDim.x + threadIdx.x;
    if (e < EE) {
        int r = ei[e], c = ei[EE + e];
        atomicAdd(&A[r * NN + c], 1.0f);
    }
}

__global__ void k_gather_x(const int* __restrict__ z, const float* __restrict__ zemb,
                           float* __restrict__ x) {
    int i = blockIdx.x * blockDim.x + threadIdx.x;
    if (i < NN * HH) {
        int n = i / HH, k = i % HH;
        x[i] = zemb[z[n] * HH + k];
    }
}

__global__ void k_transpose(const float* __restrict__ in, float* __restrict__ out,
                            int rows, int cols) {
    int i = blockIdx.x * blockDim.x + threadIdx.x;
    if (i < rows * cols) {
        int r = i / cols, c = i % cols;
        out[c * rows + r] = in[i];
    }
}

// batch is sorted; first node of each segment is src, src+1 is dst
__global__ void k_find_src(const int* __restrict__ batch, int* __restrict__ rowsIdx) {
    int n = blockIdx.x * blockDim.x + threadIdx.x;
    if (n < NN) {
        if (n == 0 || batch[n] != batch[n - 1]) {
            int g = batch[n];
            rowsIdx[g]      = n;      // src node
            rowsIdx[NG + g] = n + 1;  // dst node
        }
    }
}

__global__ void k_gather_rows(const float* __restrict__ src, float* __restrict__ dst,
                              const int* __restrict__ rowsIdx, int ncols) {
    int i = blockIdx.x * blockDim.x + threadIdx.x;
    if (i < RR * ncols) {
        int r = i / ncols, c = i % ncols;
        dst[i] = src[(size_t)rowsIdx[r] * ncols + c];
    }
}

// C[m,n] = sum_k A[m,k] * B[n,k]   (NT GEMM, fp32, one wave per 16x16 tile)
// Uses V_WMMA_F32_16X16X4_F32; M,N multiples of 16, K multiple of 4.
__global__ void k_wmma_gemm_nt(const float* __restrict__ A, const float* __restrict__ B,
                               float* __restrict__ C, int K,
                               int lda, int ldb, int ldc) {
    int lane = threadIdx.x;        // 0..31, all lanes active (EXEC all-1s for WMMA)
    int tn   = blockIdx.x * 16;
    int tm   = blockIdx.y * 16;
    int half = lane >> 4;          // 0 or 1
    int l16  = lane & 15;
    // A-frag: lane L holds row M=L%16; VGPR v holds K = v + 2*(L/16)
    const float* arow = A + (size_t)(tm + l16) * lda + 2 * half;
    const float* brow = B + (size_t)(tn + l16) * ldb + 2 * half;
    v8f acc = {};
    for (int k0 = 0; k0 < K; k0 += 4) {
        v2f af, bf;
        af[0] = arow[k0];     af[1] = arow[k0 + 1];
        bf[0] = brow[k0];     bf[1] = brow[k0 + 1];
        acc = __builtin_amdgcn_wmma_f32_16x16x4_f32(
            false, af, false, bf, (short)0, acc, false, false);
    }
    // C-frag: VGPR r -> M = r + 8*(lane/16), N = lane%16
    float* crow = C + (size_t)(tm + 8 * half) * ldc + tn + l16;
#pragma unroll
    for (int r = 0; r < 8; ++r) crow[(size_t)r * ldc] = acc[r];
}

// hid[r,z,k] = relu(f[r,z]*W1[0,k] + agg[r,z]*W1[1,k] + b1[k]),
// agg = (B[i,z] + B[z,i]) / (2*de2'), de2' = de2 + (de2 < 1)
__global__ void k_hid(const float* __restrict__ f_rows, const float* __restrict__ de2_rows,
                      const float* __restrict__ Brow, const float* __restrict__ Bcol,
                      const float* __restrict__ W1, const float* __restrict__ b1,
                      float* __restrict__ hid) {
    int rz = blockIdx.x;           // 0 .. RR*NN-1
    int k  = threadIdx.x;          // 0 .. 63
    float de2 = de2_rows[rz];
    de2 += (de2 < 1.0f) ? 1.0f : 0.0f;
    float agg = (Brow[rz] + Bcol[rz]) / (2.0f * de2);
    float h = f_rows[rz] * W1[k] + agg * W1[HH + k] + b1[k];
    hid[(size_t)rz * HH + k] = h > 0.0f ? h : 0.0f;
}

// scores[g] = hid[i,j,:]·W2[:H] + agg3[i,j,:]·W2[H:] + b2
// agg3[i,j,k] = (sum_z A[z,i] hid[j,z,k] + sum_z hid[i,z,k] A[j,z]) / de2'[i,j]
__global__ void k_pair(const float* __restrict__ At, const float* __restrict__ A,
                       const float* __restrict__ hid, const float* __restrict__ de2_rows,
                       const float* __restrict__ W2, const float* __restrict__ b2,
                       const int* __restrict__ rowsIdx, float* __restrict__ scores) {
    int g = blockIdx.x;            // pair
    int k = threadIdx.x;           // 0..63
    int i_node = rowsIdx[g];
    int j_node = rowsIdx[NG + g];
    const float* hidI = hid + (size_t)g * NN * HH;          // hid row i
    const float* hidJ = hid + (size_t)(NG + g) * NN * HH;   // hid row j
    const float* Ati  = At + (size_t)i_node * NN;           // At[i,z] = A[z,i]
    const float* Aj   = A  + (size_t)j_node * NN;           // A[j,z]
    float t = 0.0f;
    for (int z = 0; z < NN; ++z)
        t += Ati[z] * hidJ[(size_t)z * HH + k] + Aj[z] * hidI[(size_t)z * HH + k];
    float de2 = de2_rows[(size_t)g * NN + j_node];
    de2 += (de2 < 1.0f) ? 1.0f : 0.0f;
    float agg3k = t / de2;
    float hidk  = hidI[(size_t)j_node * HH + k];
    __shared__ float red[HH];
    red[k] = hidk * W2[k] + agg3k * W2[HH + k];
    __syncthreads();
    for (int s = HH / 2; s > 0; s >>= 1) {
        if (k < s) red[k] += red[k + s];
        __syncthreads();
    }
    if (k == 0) scores[g] = red[0] + b2[0];
}

// out[g] = (relu(relu(s*Wl0+bl0) @ Wl1 + bl1)) @ Wl2 + bl2
__global__ void k_mlp(const float* __restrict__ scores,
                      const float* __restrict__ Wl0, const float* __restrict__ bl0,
                      const float* __restrict__ Wl1, const float* __restrict__ bl1,
                      const float* __restrict__ Wl2, const float* __restrict__ bl2,
                      float* __restrict__ out) {
    int g = blockIdx.x;
    int h = threadIdx.x;           // 0..63
    __shared__ float o1[HH];
    __shared__ float red[HH];
    float s = scores[g];
    float v = s * Wl0[h] + bl0[h];
    o1[h] = v > 0.0f ? v : 0.0f;
    __syncthreads();
    float acc = bl1[h];
    for (int k = 0; k < HH; ++k) acc += o1[k] * Wl1[k * HH + h];
    float o2 = acc > 0.0f ? acc : 0.0f;
    red[h] = o2 * Wl2[h];
    __syncthreads();
    for (int s2 = HH / 2; s2 > 0; s2 >>= 1) {
        if (h < s2) red[h] += red[h + s2];
        __syncthreads();
    }
    if (h == 0) out[g] = red[0] + bl2[0];
}

extern "C" void kernel_launch(void* const* d_in, const int* in_sizes, int n_in,
                              void* d_out, int out_size, void* d_ws, size_t ws_size,
                              hipStream_t stream) {
    (void)in_sizes; (void)n_in; (void)out_size; (void)ws_size;
    const int*   z     = (const int*)d_in[0];
    const int*   ei    = (const int*)d_in[1];
    const int*   batch = (const int*)d_in[2];
    const float* zemb  = (const float*)d_in[3];
    const float* W1    = (const float*)d_in[4];
    const float* b1    = (const float*)d_in[5];
    const float* W2    = (const float*)d_in[6];
    const float* b2    = (const float*)d_in[7];
    const float* Wl0   = (const float*)d_in[8];
    const float* bl0   = (const float*)d_in[9];
    const float* Wl1   = (const float*)d_in[10];
    const float* bl1   = (const float*)d_in[11];
    const float* Wl2   = (const float*)d_in[12];
    const float* bl2   = (const float*)d_in[13];
    float* out = (float*)d_out;

    // workspace carve-up (~27 MB total)
    float* p      = (float*)d_ws;
    float* A      = p; p += (size_t)NN * NN;
    float* At     = p; p += (size_t)NN * NN;
    float* x      = p; p += NN * HH;
    float* xT     = p; p += HH * NN;
    float* yfull  = p; p += NN * HH;   // Aᵀ x
    float* x_rows = p; p += RR * HH;
    float* A_rows = p; p += (size_t)RR * NN;
    float* y_rows = p; p += RR * HH;
    float* f_rows = p; p += (size_t)RR * NN;
    float* de2_r  = p; p += (size_t)RR * NN;
    float* Brow   = p; p += (size_t)RR * NN;   // B[i,z]
    float* Bcol   = p; p += (size_t)RR * NN;   // B[z,i]
    float* hid    = p; p += (size_t)RR * NN * HH;  // 16 MB
    float* scores = p; p += NG;
    int*   rowsIdx = (int*)p;

    k_zero<<<(NN * NN + 255) / 256, 256, 0, stream>>>(A, NN * NN);
    k_build_A<<<(EE + 255) / 256, 256, 0, stream>>>(ei, A);
    k_gather_x<<<(NN * HH + 255) / 256, 256, 0, stream>>>(z, zemb, x);
    k_transpose<<<(NN * NN + 255) / 256, 256, 0, stream>>>(A, At, NN, NN);
    k_transpose<<<(NN * HH + 255) / 256, 256, 0, stream>>>(x, xT, NN, HH);
    k_find_src<<<(NN + 255) / 256, 256, 0, stream>>>(batch, rowsIdx);

    // yfull[z,k] = sum_w At[z,w] * xT[k,w]   (M=NN, N=HH, K=NN)
    { dim3 grid(HH / 16, NN / 16);
      k_wmma_gemm_nt<<<grid, 32, 0, stream>>>(At, xT, yfull, NN, NN, NN, HH); }

    k_gather_rows<<<(RR * HH + 255) / 256, 256, 0, stream>>>(x, x_rows, rowsIdx, HH);
    k_gather_rows<<<(RR * NN + 255) / 256, 256, 0, stream>>>(A, A_rows, rowsIdx, NN);
    k_gather_rows<<<(RR * HH + 255) / 256, 256, 0, stream>>>(yfull, y_rows, rowsIdx, HH);

    dim3 grid64(NN / 16, RR / 16);
    // f_rows[r,z]  = x_rows[r,:]·x[z,:]        (K=64)
    k_wmma_gemm_nt<<<grid64, 32, 0, stream>>>(x_rows, x, f_rows, HH, HH, HH, NN);
    // de2_r[r,z]   = A_rows[r,:]·A[z,:]        (K=1024)
    k_wmma_gemm_nt<<<grid64, 32, 0, stream>>>(A_rows, A, de2_r, NN, NN, NN, NN);
    // Brow[r,z]    = y_rows[r,:]·x[z,:]  = B[i,z]
    k_wmma_gemm_nt<<<grid64, 32, 0, stream>>>(y_rows, x, Brow, HH, HH, HH, NN);
    // Bcol[r,z]    = x_rows[r,:]·yfull[z,:] = B[z,i]
    k_wmma_gemm_nt<<<grid64, 32, 0, stream>>>(x_rows, yfull, Bcol, HH, HH, HH, NN);

    k_hid<<<RR * NN, HH, 0, stream>>>(f_rows, de2_r, Brow, Bcol, W1, b1, hid);
    k_pair<<<NG, HH, 0, stream>>>(At, A, hid, de2_r, W2, b2, rowsIdx, scores);
    k_mlp<<<NG, HH, 0, stream>>>(scores, Wl0, bl0, Wl1, bl1, Wl2, bl2, out);
}